// CuboidLoss_1700807049809
// MI455X (gfx1250) — compile-verified
//
#include <hip/hip_runtime.h>
#include <math.h>

#define B_ 64
#define P_ 8396
#define K_ 6
#define C_ 25
#define TILE 256
#define TPB ((P_ + TILE - 1) / TILE) /* 33 */
#define NPART 1024
#define MT 1024
#define ITER ((P_ + MT - 1) / MT) /* 9 */
#define KEY_NEGINF 0x007FFFFFu

typedef __attribute__((ext_vector_type(2))) float v2f;
typedef __attribute__((ext_vector_type(8))) float v8f;

__device__ __forceinline__ unsigned ordkey(float f) {
  unsigned u = __float_as_uint(f);
  return (u & 0x80000000u) ? ~u : (u | 0x80000000u);
}

// ---------------- init ----------------
__global__ void init_ws(unsigned long long* best, int* npos, float* part, int nPartTot) {
  int i = blockIdx.x * blockDim.x + threadIdx.x;
  if (i < B_) { best[i] = 0ull; npos[i] = 0; }
  if (i < nPartTot) part[i] = 0.f;
}

// ---------------- IoU + argmax ----------------
__global__ __launch_bounds__(256) void iou_kernel(const float* __restrict__ prior,
                                                  const float* __restrict__ gt,
                                                  float* __restrict__ iou,
                                                  unsigned long long* __restrict__ best) {
  __shared__ float sgt[25];
  __shared__ unsigned long long skey[256];
  int b = blockIdx.x / TPB;
  int tile = blockIdx.x % TPB;
  int tid = threadIdx.x;
  if (tid < 25) sgt[tid] = gt[b * 25 + tid];
  __syncthreads();
  int p = tile * TILE + tid;
  float iouv = 0.f;
  if (p < P_) {
    const float4* pr4 = (const float4*)(prior + (size_t)p * 24);
    float cs = 0.f, acc = 0.f;
#pragma unroll
    for (int k = 0; k < K_; ++k) {
      float4 pk = pr4[k];
      float g0 = sgt[1 + 4 * k], g1 = sgt[2 + 4 * k];
      float g2 = sgt[3 + 4 * k], g3 = sgt[4 + 4 * k];
      float ix0 = fmaxf(pk.x, g0), iy0 = fmaxf(pk.y, g1);
      float ix1 = fminf(pk.z, g2), iy1 = fminf(pk.w, g3);
      float cw = fmaxf(ix1 - ix0, 0.f), ch = fmaxf(iy1 - iy0, 0.f);
      float cr = cw * ch;
      float pa = (pk.z - pk.x) * (pk.w - pk.y);
      float ga = (g2 - g0) * (g3 - g1);
      cs += cr;
      acc += cr / (pa + ga - cr);
    }
    iouv = (cs > 0.f) ? acc * (1.0f / (float)K_) : 0.f;
    iou[(size_t)b * P_ + p] = iouv;
  }
  unsigned long long key = 0ull;
  if (p < P_)
    key = ((unsigned long long)__float_as_uint(iouv) << 32) |
          (unsigned long long)(0xFFFFFFFFu - (unsigned)p);
  skey[tid] = key;
  __syncthreads();
  for (int s = 128; s > 0; s >>= 1) {
    if (tid < s) { if (skey[tid + s] > skey[tid]) skey[tid] = skey[tid + s]; }
    __syncthreads();
  }
  if (tid == 0) atomicMax(&best[b], skey[0]);
}

// ---------------- softmax / CE / loc loss ----------------
__global__ __launch_bounds__(256) void conf_loc_kernel(
    const float* __restrict__ loc, const float* __restrict__ conf,
    const float* __restrict__ prior, const float* __restrict__ gt,
    const float* __restrict__ iou, const unsigned long long* __restrict__ best,
    float* __restrict__ neg_score, float* __restrict__ ce_out,
    int* __restrict__ npos, float* __restrict__ part) {
  __shared__ float sgt[25];
  __shared__ float rl[256], rc[256];
  __shared__ int rn[256];
  int b = blockIdx.x / TPB;
  int tile = blockIdx.x % TPB;
  int tid = threadIdx.x;
  if (tid < 25) sgt[tid] = gt[b * 25 + tid];
  __syncthreads();
  int p = tile * TILE + tid;
  float ll = 0.f, cep = 0.f;
  int np = 0;
  if (p < P_) {
    size_t idx = (size_t)b * P_ + p;
    unsigned bestp = 0xFFFFFFFFu - (unsigned)(best[b] & 0xFFFFFFFFull);
    bool pos = (iou[idx] >= 0.5f) || ((unsigned)p == bestp);
    int target = pos ? (int)sgt[0] : 0;
    const float* cr = conf + idx * C_;
    if (tile + 1 < TPB) __builtin_prefetch(cr + TILE * C_, 0, 0); // global_prefetch_b8
    float l[C_];
    float m = -__builtin_inff();
    float lt = 0.f;
#pragma unroll
    for (int j = 0; j < C_; ++j) {
      l[j] = cr[j];
      m = fmaxf(m, l[j]);
      lt = (j == target) ? l[j] : lt;
    }
    float se = 0.f;
#pragma unroll
    for (int j = 0; j < C_; ++j) se += expf(l[j] - m);
    float logZ = m + logf(se);
    float ce = logZ - lt;
    float pt = expf(lt - logZ);
    float tl = -logf(pt + 1e-6f);
    ce_out[idx] = ce;
    neg_score[idx] = pos ? -__builtin_inff() : tl;
    if (pos) {
      np = 1;
      cep = ce;
      const float4* pr4 = (const float4*)(prior + (size_t)p * 24);
      const float4* lr4 = (const float4*)(loc + idx * 24);
#pragma unroll
      for (int k = 0; k < K_; ++k) {
        float4 pk = pr4[k];
        float4 lk = lr4[k];
        float g0 = sgt[1 + 4 * k], g1 = sgt[2 + 4 * k];
        float g2 = sgt[3 + 4 * k], g3 = sgt[4 + 4 * k];
        float pcx = (pk.x + pk.z) * 0.5f, pcy = (pk.y + pk.w) * 0.5f;
        float pw = pk.z - pk.x, ph = pk.w - pk.y;
        float gcx = (g0 + g2) * 0.5f, gcy = (g1 + g3) * 0.5f;
        float gw = g2 - g0, gh = g3 - g1;
        float e0 = (gcx - pcx) / pw * 10.f;
        float e1 = (gcy - pcy) / ph * 10.f;
        float e2 = logf(gw / pw) * 5.f;
        float e3 = logf(gh / ph) * 5.f;
        float d;
        d = fabsf(lk.x - e0); ll += (d < 1.f) ? 0.5f * d * d : d - 0.5f;
        d = fabsf(lk.y - e1); ll += (d < 1.f) ? 0.5f * d * d : d - 0.5f;
        d = fabsf(lk.z - e2); ll += (d < 1.f) ? 0.5f * d * d : d - 0.5f;
        d = fabsf(lk.w - e3); ll += (d < 1.f) ? 0.5f * d * d : d - 0.5f;
      }
    }
  }
  rl[tid] = ll; rc[tid] = cep; rn[tid] = np;
  __syncthreads();
  for (int s = 128; s > 0; s >>= 1) {
    if (tid < s) { rl[tid] += rl[tid + s]; rc[tid] += rc[tid + s]; rn[tid] += rn[tid + s]; }
    __syncthreads();
  }
  if (tid == 0) {
    int slot = (int)(blockIdx.x & (NPART - 1));
    atomicAdd(&part[0 * NPART + slot], rl[0]);          // smooth-L1 sum
    atomicAdd(&part[1 * NPART + slot], rc[0]);          // ce over positives
    atomicAdd(&part[2 * NPART + slot], (float)rn[0]);   // N (pos count)
    atomicAdd(&part[3 * NPART + slot], (float)rn[0]);   // sel count (pos part)
    if (rn[0]) atomicAdd(&npos[b], rn[0]);
  }
}

// ---------------- hard negative mining: per-row radix top-k ----------------
__global__ __launch_bounds__(MT) void mining_kernel(const float* __restrict__ neg_score,
                                                    const float* __restrict__ ce,
                                                    const int* __restrict__ npos,
                                                    float* __restrict__ part) {
  __shared__ unsigned skeys[P_];
  __shared__ unsigned hist[256];
  __shared__ unsigned s_prefix, s_kk, eqcnt;
  __shared__ float red[MT];
  int b = blockIdx.x, tid = threadIdx.x;
  int npb = npos[b];
  int numneg = P_ - npb;
  int kk = 3 * npb;
  if (kk > numneg) kk = numneg;
  for (int i = 0; i < ITER; ++i) {
    int p = tid + i * MT;
    if (p < P_) skeys[p] = ordkey(neg_score[(size_t)b * P_ + p]); // -inf -> KEY_NEGINF
  }
  if (tid == 0) { s_prefix = 0u; s_kk = (unsigned)kk; eqcnt = 0u; }
  __syncthreads();
  if (kk > 0) {
    for (int pass = 0; pass < 4; ++pass) {
      unsigned shift = 24u - (unsigned)pass * 8u;
      unsigned himask = (pass == 0) ? 0u : (0xFFFFFFFFu << (32 - 8 * pass));
      unsigned prefix = s_prefix;
      if (tid < 256) hist[tid] = 0u;
      __syncthreads();
      for (int i = 0; i < ITER; ++i) {
        int p = tid + i * MT;
        if (p < P_) {
          unsigned key = skeys[p];
          if (key > KEY_NEGINF && (key & himask) == prefix)
            atomicAdd(&hist[(key >> shift) & 255u], 1u);
        }
      }
      __syncthreads();
      if (tid == 0) {
        unsigned run = 0u, need = s_kk, dsel = 0u;
        for (int d = 255; d >= 0; --d) {
          unsigned c = hist[d];
          if (run + c >= need) { dsel = (unsigned)d; s_kk = need - run; break; }
          run += c;
        }
        s_prefix = prefix | (dsel << shift);
      }
      __syncthreads();
    }
    unsigned T = s_prefix;
    unsigned kkrem = s_kk;
    float sum = 0.f;
    for (int i = 0; i < ITER; ++i) {
      int p = tid + i * MT;
      if (p < P_) {
        unsigned key = skeys[p];
        if (key > KEY_NEGINF) {
          if (key > T) {
            sum += ce[(size_t)b * P_ + p];
          } else if (key == T) {
            unsigned slot = atomicAdd(&eqcnt, 1u);
            if (slot < kkrem) sum += ce[(size_t)b * P_ + p]; // ties have equal ce
          }
        }
      }
    }
    red[tid] = sum;
    __syncthreads();
    for (int s = MT / 2; s > 0; s >>= 1) {
      if (tid < s) red[tid] += red[tid + s];
      __syncthreads();
    }
    if (tid == 0) {
      int slot = b & (NPART - 1);
      atomicAdd(&part[1 * NPART + slot], red[0]);     // ce over mined negatives
      atomicAdd(&part[3 * NPART + slot], (float)kk);  // sel count (neg part)
    }
  }
}

// ---------------- WMMA partial-sum reduction + finalize ----------------
// Sums each NPART-float array with V_WMMA_F32_16X16X4_F32: D = A(16x4) * ones(4x16) + C.
// Every D element is a row-sum replicated across 16 columns -> total = sum(D)/16.
__global__ __launch_bounds__(32) void finalize_kernel(const float* __restrict__ part,
                                                      float* __restrict__ out) {
  __shared__ float red[4][32];
  int lane = threadIdx.x;
  v2f ones; ones[0] = 1.f; ones[1] = 1.f;
  v8f acc0 = {}, acc1 = {}, acc2 = {}, acc3 = {};
  for (int i = 0; i < NPART / 64; ++i) {
    int base = i * 64 + lane * 2;
    v2f a0, a1, a2, a3;
    a0[0] = part[0 * NPART + base]; a0[1] = part[0 * NPART + base + 1];
    a1[0] = part[1 * NPART + base]; a1[1] = part[1 * NPART + base + 1];
    a2[0] = part[2 * NPART + base]; a2[1] = part[2 * NPART + base + 1];
    a3[0] = part[3 * NPART + base]; a3[1] = part[3 * NPART + base + 1];
    acc0 = __builtin_amdgcn_wmma_f32_16x16x4_f32(false, a0, false, ones, (short)0, acc0, false, false);
    acc1 = __builtin_amdgcn_wmma_f32_16x16x4_f32(false, a1, false, ones, (short)0, acc1, false, false);
    acc2 = __builtin_amdgcn_wmma_f32_16x16x4_f32(false, a2, false, ones, (short)0, acc2, false, false);
    acc3 = __builtin_amdgcn_wmma_f32_16x16x4_f32(false, a3, false, ones, (short)0, acc3, false, false);
  }
  float s0 = 0.f, s1 = 0.f, s2 = 0.f, s3 = 0.f;
#pragma unroll
  for (int j = 0; j < 8; ++j) { s0 += acc0[j]; s1 += acc1[j]; s2 += acc2[j]; s3 += acc3[j]; }
  red[0][lane] = s0; red[1][lane] = s1; red[2][lane] = s2; red[3][lane] = s3;
  __syncthreads();
  if (lane == 0) {
    float t[4];
    for (int q = 0; q < 4; ++q) {
      float s = 0.f;
      for (int j = 0; j < 32; ++j) s += red[q][j];
      t[q] = s * (1.0f / 16.0f);
    }
    float N = t[2];
    out[0] = (N > 0.f) ? t[0] / (float)K_ / N : 0.f;
    out[1] = (t[3] > 0.f) ? t[1] / t[3] : 0.f;
  }
}

extern "C" void kernel_launch(void* const* d_in, const int* in_sizes, int n_in,
                              void* d_out, int out_size, void* d_ws, size_t ws_size,
                              hipStream_t stream) {
  (void)in_sizes; (void)n_in; (void)out_size; (void)ws_size;
  const float* loc = (const float*)d_in[0];
  const float* conf = (const float*)d_in[1];
  const float* prior = (const float*)d_in[2];
  const float* gt = (const float*)d_in[3];
  float* out = (float*)d_out;

  size_t BP = (size_t)B_ * P_;
  float* iou = (float*)d_ws;
  float* ns = iou + BP;
  float* cearr = ns + BP;
  unsigned long long* best = (unsigned long long*)(cearr + BP); // 8B aligned
  int* npos = (int*)(best + B_);
  float* part = (float*)(npos + B_); // 4 * NPART floats

  init_ws<<<dim3((4 * NPART + 255) / 256), dim3(256), 0, stream>>>(best, npos, part, 4 * NPART);
  dim3 grid(B_ * TPB), blk(256);
  iou_kernel<<<grid, blk, 0, stream>>>(prior, gt, iou, best);
  conf_loc_kernel<<<grid, blk, 0, stream>>>(loc, conf, prior, gt, iou, best, ns, cearr, npos, part);
  mining_kernel<<<dim3(B_), dim3(MT), 0, stream>>>(ns, cearr, npos, part);
  finalize_kernel<<<dim3(1), dim3(32), 0, stream>>>(part, out);
}